// PercentileBasedDFDetector_67164698575283
// MI455X (gfx1250) — compile-verified
//
#include <hip/hip_runtime.h>
#include <math.h>

// ---------------- problem constants ----------------
#define B_      8
#define C_      32
#define H_      512
#define W_      512
#define DW      256                // downsampled width/height
#define N_PIX   (H_ * W_)          // 262144 per batch
#define MT      8                  // down-rows per band (16 output rows)
#define BANDS   (DW / MT)          // 32 bands per batch
#define NT      256                // threads per block
#define SROWS   (2 * MT + 4)       // 20 staged x rows per band

struct Bounds { int r[16]; };      // ascending rank boundaries, r[15] == k

// ---- CDNA5 async global->LDS path (guarded; falls back to plain LDS staging) ----
#if __has_builtin(__builtin_amdgcn_global_load_async_to_lds_b128) && \
    __has_builtin(__builtin_amdgcn_s_wait_asynccnt)
#define USE_ASYNC_LDS 1
typedef int v4i __attribute__((ext_vector_type(4)));
typedef __attribute__((address_space(1))) v4i as1_v4i;   // global
typedef __attribute__((address_space(3))) v4i as3_v4i;   // LDS
#else
#define USE_ASYNC_LDS 0
#endif

// =====================================================================
// Kernel A: df = sum_c |x - up(down(x))|
//   down: 2x2 average pool; up: separable [0.25,0.75] half-pixel lerp,
//   edge-clamped (identical to jax bilinear, antialias=False).
// One block: batch b, 16 output rows (8 down rows), full 512 width.
// Thread t owns down-column w=t -> output columns 2t, 2t+1.
// Channel loop is double-buffered: async global->LDS DMA for channel
// ch+1 overlaps the stencil compute of channel ch (ASYNCcnt pipeline).
// =====================================================================
__global__ __launch_bounds__(NT) void df_kernel(const float* __restrict__ x,
                                                float* __restrict__ df) {
  const int blk  = blockIdx.x;
  const int b    = blk / BANDS;
  const int band = blk % BANDS;
  const int m0   = band * MT;
  const int t    = threadIdx.x;
  const int w    = t;                       // down column 0..255

  __shared__ float xt[2][SROWS][W_];        // 2 x 20 x 512 staged rows (80KB)
  __shared__ float dt[MT + 2][DW + 1];      // 10 x 257 down rows (pad)

  // precompute clamped source row for each staged row slot (band-constant)
  int hrow[SROWS];
#pragma unroll
  for (int rr = 0; rr < SROWS; ++rr) {
    int m = m0 - 1 + (rr >> 1);
    m = m < 0 ? 0 : (m > DW - 1 ? DW - 1 : m);
    hrow[rr] = 2 * m + (rr & 1);
  }

  float acc[2 * MT][2];
#pragma unroll
  for (int i = 0; i < 2 * MT; ++i) { acc[i][0] = 0.0f; acc[i][1] = 0.0f; }

  const size_t imgBase = (size_t)b * C_ * H_ * W_;

  // ---- stage channel 0 into buffer 0 ----
  {
    const float* chp = x + imgBase;         // channel 0
#pragma unroll
    for (int it = 0; it < (SROWS * (W_ / 4)) / NT; ++it) {
      const int chunk = t + it * NT;
      const int rr = chunk >> 7;            // 128 float4 per row
      const int cv = chunk & 127;
      const float* gp = chp + (size_t)hrow[rr] * W_ + (size_t)(cv * 4);
#if USE_ASYNC_LDS
      __builtin_amdgcn_global_load_async_to_lds_b128(
          (as1_v4i*)gp, (as3_v4i*)&xt[0][rr][cv * 4], 0, 0);
#else
      *(float4*)&xt[0][rr][cv * 4] = *(const float4*)gp;
#endif
    }
  }

  for (int ch = 0; ch < C_; ++ch) {
    const int cur = ch & 1;
#if USE_ASYNC_LDS
    __builtin_amdgcn_s_wait_asynccnt(0);    // buffer `cur` fully landed (this wave)
#endif
    __syncthreads();                        // ...and all other waves' chunks too

    // ---- issue async staging of channel ch+1 into the other buffer ----
    if (ch + 1 < C_) {
      const float* chp = x + imgBase + (size_t)(ch + 1) * H_ * W_;
#pragma unroll
      for (int it = 0; it < (SROWS * (W_ / 4)) / NT; ++it) {
        const int chunk = t + it * NT;
        const int rr = chunk >> 7;
        const int cv = chunk & 127;
        const float* gp = chp + (size_t)hrow[rr] * W_ + (size_t)(cv * 4);
#if USE_ASYNC_LDS
        __builtin_amdgcn_global_load_async_to_lds_b128(
            (as1_v4i*)gp, (as3_v4i*)&xt[cur ^ 1][rr][cv * 4], 0, 0);
#else
        *(float4*)&xt[cur ^ 1][rr][cv * 4] = *(const float4*)gp;
#endif
      }
    }

    // ---- down rows: 2x2 average (reads buffer `cur`) ----
#pragma unroll
    for (int rl = 0; rl < MT + 2; ++rl) {
      const float s = xt[cur][2 * rl][2 * w] + xt[cur][2 * rl][2 * w + 1] +
                      xt[cur][2 * rl + 1][2 * w] + xt[cur][2 * rl + 1][2 * w + 1];
      dt[rl][w] = 0.25f * s;
    }
    __syncthreads();

    // ---- recon + accumulate |x - recon| ----
    const int wl = (w == 0) ? 0 : w - 1;
    const int wr = (w == DW - 1) ? DW - 1 : w + 1;
#pragma unroll
    for (int rl = 1; rl <= MT; ++rl) {
      const float dAc = dt[rl - 1][w], dAl = dt[rl - 1][wl], dAr = dt[rl - 1][wr];
      const float dBc = dt[rl][w],     dBl = dt[rl][wl],     dBr = dt[rl][wr];
      const float dCc = dt[rl + 1][w], dCl = dt[rl + 1][wl], dCr = dt[rl + 1][wr];
      // horizontal lerps for out cols 2w (left-neighbor) and 2w+1 (right-neighbor)
      const float hA0 = 0.75f * dAc + 0.25f * dAl, hA1 = 0.75f * dAc + 0.25f * dAr;
      const float hB0 = 0.75f * dBc + 0.25f * dBl, hB1 = 0.75f * dBc + 0.25f * dBr;
      const float hC0 = 0.75f * dCc + 0.25f * dCl, hC1 = 0.75f * dCc + 0.25f * dCr;
      // vertical: row 2m uses (m-1,m)@(0.25,0.75); row 2m+1 uses (m,m+1)@(0.75,0.25)
      const float rT0 = 0.25f * hA0 + 0.75f * hB0;
      const float rT1 = 0.25f * hA1 + 0.75f * hB1;
      const float rU0 = 0.75f * hB0 + 0.25f * hC0;
      const float rU1 = 0.75f * hB1 + 0.25f * hC1;
      const float x00 = xt[cur][2 * rl][2 * w],     x01 = xt[cur][2 * rl][2 * w + 1];
      const float x10 = xt[cur][2 * rl + 1][2 * w], x11 = xt[cur][2 * rl + 1][2 * w + 1];
      acc[2 * (rl - 1)][0]     += fabsf(x00 - rT0);
      acc[2 * (rl - 1)][1]     += fabsf(x01 - rT1);
      acc[2 * (rl - 1) + 1][0] += fabsf(x10 - rU0);
      acc[2 * (rl - 1) + 1][1] += fabsf(x11 - rU1);
    }
    // no trailing barrier needed: next iteration's wait+syncthreads orders
    // dt/xt reuse, and the in-flight async loads target the other buffer.
  }

  // ---- write df: 16 rows x (cols 2t,2t+1) as float2 (coalesced) ----
  float* dfb = df + (size_t)b * N_PIX + (size_t)(2 * m0) * W_ + (size_t)(2 * w);
#pragma unroll
  for (int i = 0; i < 2 * MT; ++i) {
    *(float2*)(dfb + (size_t)i * W_) = make_float2(acc[i][0], acc[i][1]);
  }
}

// =====================================================================
// Kernel B: per (batch, boundary j) 3-level radix select over df bits.
// Finds V = key at descending 0-indexed rank (r[j]-1) and
// G = count(key > V). df >= 0 so uint bit order == value order.
// df is L2-resident (1MB/batch, 192MB L2), so the 3 passes are cheap.
// =====================================================================
__global__ __launch_bounds__(256) void select_kernel(const unsigned* __restrict__ keys,
                                                     unsigned* __restrict__ Vout,
                                                     unsigned* __restrict__ Gout,
                                                     Bounds bd) {
  const int batch = blockIdx.x >> 4;
  const int j     = blockIdx.x & 15;
  const unsigned* bk = keys + (size_t)batch * N_PIX;
  const unsigned target = (unsigned)bd.r[j] - 1u;
  const int t = threadIdx.x;

  __shared__ unsigned hist[2048];
  __shared__ unsigned s_bin, s_rem;

  // pass 1: bits[31:21]
  for (int i = t; i < 2048; i += 256) hist[i] = 0u;
  __syncthreads();
  for (int i = t; i < N_PIX; i += 256) atomicAdd(&hist[bk[i] >> 21], 1u);
  __syncthreads();
  if (t == 0) {
    unsigned rem = target; int bin = 2047;
    for (;;) { const unsigned c = hist[bin]; if (c > rem || bin == 0) break; rem -= c; --bin; }
    s_bin = (unsigned)bin; s_rem = rem;
  }
  __syncthreads();
  const unsigned c1 = s_bin;
  const unsigned rem1 = s_rem;

  // pass 2: bits[20:10] among keys whose top bits == c1
  __syncthreads();
  for (int i = t; i < 2048; i += 256) hist[i] = 0u;
  __syncthreads();
  for (int i = t; i < N_PIX; i += 256) {
    const unsigned kk = bk[i];
    if ((kk >> 21) == c1) atomicAdd(&hist[(kk >> 10) & 0x7FFu], 1u);
  }
  __syncthreads();
  if (t == 0) {
    unsigned rem = rem1; int bin = 2047;
    for (;;) { const unsigned c = hist[bin]; if (c > rem || bin == 0) break; rem -= c; --bin; }
    s_bin = (unsigned)bin; s_rem = rem;
  }
  __syncthreads();
  const unsigned pref = (c1 << 11) | s_bin;
  const unsigned rem2 = s_rem;

  // pass 3: bits[9:0] among keys whose bits[31:10] == pref
  __syncthreads();
  for (int i = t; i < 1024; i += 256) hist[i] = 0u;
  __syncthreads();
  for (int i = t; i < N_PIX; i += 256) {
    const unsigned kk = bk[i];
    if ((kk >> 10) == pref) atomicAdd(&hist[kk & 0x3FFu], 1u);
  }
  __syncthreads();
  if (t == 0) {
    unsigned rem = rem2; int bin = 1023;
    for (;;) { const unsigned c = hist[bin]; if (c > rem || bin == 0) break; rem -= c; --bin; }
    Vout[batch * 16 + j] = (pref << 10) | (unsigned)bin;
    Gout[batch * 16 + j] = target - rem;   // count strictly greater than V
  }
}

// =====================================================================
// Kernel C: conn = max(1, #{ boundaries j : rank(pixel) < r[j] }).
//   key >  V_j  -> beats boundary j
//   key == V_j  -> resolve deterministically by index order among equals
// batch is derived from blockIdx (1024 blocks per batch) so V/G loads
// are wave-uniform -> scalar loads.
// =====================================================================
__global__ __launch_bounds__(256) void conn_kernel(const unsigned* __restrict__ keys,
                                                   const unsigned* __restrict__ Vg,
                                                   const unsigned* __restrict__ Gg,
                                                   float* __restrict__ out,
                                                   Bounds bd) {
  const unsigned batch = blockIdx.x >> 10;                     // 1024 blocks/batch
  const unsigned pix   = ((blockIdx.x & 1023u) << 8) + threadIdx.x;
  const unsigned* bk = keys + (size_t)batch * N_PIX;
  const unsigned key = bk[pix];

  int cnt = 0;
  unsigned tiemask = 0u;
#pragma unroll
  for (int j = 0; j < 16; ++j) {
    const unsigned v = Vg[batch * 16 + j];
    cnt += (key > v) ? 1 : 0;
    tiemask |= (key == v) ? (1u << j) : 0u;
  }
  if (tiemask) {
    // exact rank among equal keys: # equal keys with smaller pixel index
    unsigned eqr = 0;
    for (unsigned i = 0; i < pix; ++i) eqr += (bk[i] == key) ? 1u : 0u;
#pragma unroll
    for (int j = 0; j < 16; ++j) {
      if (tiemask & (1u << j)) {
        const unsigned rank = Gg[batch * 16 + j] + eqr;
        if (rank < (unsigned)bd.r[j]) ++cnt;
      }
    }
  }
  out[(size_t)batch * N_PIX + pix] = (float)(cnt > 0 ? cnt : 1);  // MIN_CONN floor
}

// =====================================================================
extern "C" void kernel_launch(void* const* d_in, const int* in_sizes, int n_in,
                              void* d_out, int out_size, void* d_ws, size_t ws_size,
                              hipStream_t stream) {
  (void)in_sizes; (void)n_in; (void)out_size; (void)ws_size;
  const float* x = (const float*)d_in[0];
  float* out  = (float*)d_out;
  float* df   = out;                               // 8*262144 floats
  float* conn = out + (size_t)B_ * N_PIX;          // second output
  unsigned* Vg = (unsigned*)d_ws;                  // 128 entries
  unsigned* Gg = Vg + 128;                         // 128 entries

  // rank boundaries: conn bucket v occupied for i in [c_{v+1}, c_v),
  // c_v = first i with 15*(1-i/k)^2 < v-0.5  (never exactly half -> exact)
  Bounds bd;
  const int k = (N_PIX * 10) / 100;                // 26214, matches int(hw*10/100)
  for (int v = 15; v >= 1; --v) {
    const double s = sqrt(((double)v - 0.5) / 15.0);
    bd.r[15 - v] = (int)floor((double)k * (1.0 - s)) + 1;
  }
  bd.r[15] = k;

  df_kernel<<<dim3(B_ * BANDS), dim3(NT), 0, stream>>>(x, df);
  select_kernel<<<dim3(B_ * 16), dim3(256), 0, stream>>>((const unsigned*)df, Vg, Gg, bd);
  conn_kernel<<<dim3((B_ * N_PIX) / 256), dim3(256), 0, stream>>>((const unsigned*)df, Vg, Gg, conn, bd);
}